// ComputeLoss_EIOU_17360257811110
// MI455X (gfx1250) — compile-verified
//
#include <hip/hip_runtime.h>
#include <math.h>

// Problem constants (from reference): B=4, T=256, A=3, C=13, F=18, GRID=64, SCALE=4
#define GRID_N 64
#define BS 4
#define TT 256
#define ANCH 3
#define CC 13
#define FF 18
#define NTGT (BS*TT)                               // 1024
#define NCAND (7*ANCH*NTGT)                        // 21504
#define NCELLS (BS*ANCH*GRID_N*GRID_N*GRID_N)      // 3,145,728
#define NMASKW (NCELLS/32)                         // 98,304 u32 words (384 KB)

typedef __attribute__((ext_vector_type(2))) float v2f;
typedef __attribute__((ext_vector_type(8))) float v8f;

__device__ __forceinline__ float softplusf(float x) {
    // stable: max(x,0) + log1p(exp(-|x|))
    return fmaxf(x, 0.0f) + log1pf(expf(-fabsf(x)));
}
__device__ __forceinline__ float sigmoidf(float x) {
    return 1.0f / (1.0f + expf(-x));
}

// Workspace layout: float acc[16] at offset 0 (0:sum_w 1:sum_bbox 2:sum_cls 3:sum_obj),
// then unsigned mask[NMASKW] at offset 64.
__global__ void k_zero(float* acc, unsigned* mask) {
    int i = blockIdx.x * blockDim.x + threadIdx.x;
    if (i < NMASKW) mask[i] = 0u;
    if (i < 16)     acc[i]  = 0.0f;
}

__global__ void k_candidates(const float* __restrict__ p,
                             const float* __restrict__ targets,
                             const float* __restrict__ anchor,
                             float* acc, unsigned* mask) {
    int id = blockIdx.x * blockDim.x + threadIdx.x;
    if (id >= NCAND) return;
    int t = id % NTGT;
    int a = (id / NTGT) % ANCH;
    int o = id / (NTGT * ANCH);
    int b = t / TT;

    const float* x = targets + (long)t * FF;
    if (!(x[4] > 0.5f)) return;                        // conf_m

    float rn = x[3] * 0.25f;                           // r / SCALE
    float an = anchor[a] * 0.25f;                      // anchor_norm
    float ratio = rn / an;
    if (fmaxf(ratio, 1.0f / ratio) >= 4.0f) return;    // aok

    float gx = x[0] * 0.25f, gy = x[1] * 0.25f, gz = x[2] * 0.25f;

    const float g = 0.5f;
    bool fm = true;
    float ox = 0.0f, oy = 0.0f, oz = 0.0f;
    switch (o) {
        case 0: break;
        case 1: fm = ((gx - truncf(gx)) < g) && (gx > 1.0f); ox =  g; break;
        case 2: fm = ((gy - truncf(gy)) < g) && (gy > 1.0f); oy =  g; break;
        case 3: fm = ((gz - truncf(gz)) < g) && (gz > 1.0f); oz =  g; break;
        case 4: { float q = 64.0f - gx; fm = ((q - truncf(q)) < g) && (q > 1.0f); ox = -g; } break;
        case 5: { float q = 64.0f - gy; fm = ((q - truncf(q)) < g) && (q > 1.0f); oy = -g; } break;
        case 6: { float q = 64.0f - gz; fm = ((q - truncf(q)) < g) && (q > 1.0f); oz = -g; } break;
    }
    if (!fm) return;

    // gijk_f = trunc(gxyz - off); indices clipped, tbox uses unclipped gijk_f
    float fx = truncf(gx - ox), fy = truncf(gy - oy), fz = truncf(gz - oz);
    int gi = min(max((int)fx, 0), GRID_N - 1);
    int gj = min(max((int)fy, 0), GRID_N - 1);
    int gk = min(max((int)fz, 0), GRID_N - 1);

    long lin = ((((long)b * ANCH + a) * GRID_N + gk) * GRID_N + gj) * GRID_N + gi;
    const float* pr = p + lin * FF;

    // pbox
    float c1x = sigmoidf(pr[0]) * 2.0f - 0.5f;
    float c1y = sigmoidf(pr[1]) * 2.0f - 0.5f;
    float c1z = sigmoidf(pr[2]) * 2.0f - 0.5f;
    float s3  = sigmoidf(pr[3]) * 2.0f;
    float r1  = s3 * s3 * an;
    // tbox
    float c2x = gx - fx, c2y = gy - fy, c2z = gz - fz;
    float r2  = rn;

    const float eps = 1e-7f;
    float h1 = r1 * 0.5f, h2 = r2 * 0.5f;
    float c1v[3] = {c1x, c1y, c1z};
    float c2v[3] = {c2x, c2y, c2z};
    float dr2 = (r1 - r2) * (r1 - r2);
    float inter = 1.0f, rho2 = 0.0f, cdiag = 0.0f, spen = 0.0f;
    #pragma unroll
    for (int d = 0; d < 3; ++d) {
        float lo_ = fmaxf(c1v[d] - h1, c2v[d] - h2);
        float hi_ = fminf(c1v[d] + h1, c2v[d] + h2);
        inter *= fmaxf(hi_ - lo_, 0.0f);
        float clo = fminf(c1v[d] - h1, c2v[d] - h2);
        float chi = fmaxf(c1v[d] + h1, c2v[d] + h2);
        float cd  = chi - clo;
        float dd  = c1v[d] - c2v[d];
        rho2  += dd * dd;
        cdiag += cd * cd;
        spen  += dr2 / (cd * cd + eps);
    }
    float uni = r1 * r1 * r1 + r2 * r2 * r2 - inter + eps;
    float ei  = inter / uni - rho2 / (cdiag + eps) - spen;

    // class BCE against one-hot target
    float cls = 0.0f;
    #pragma unroll
    for (int c = 0; c < CC; ++c) {
        float l  = pr[5 + c];
        float tc = x[5 + c];
        cls += tc * softplusf(-l) + (1.0f - tc) * softplusf(l);
    }

    atomicAdd(&acc[0], 1.0f);
    atomicAdd(&acc[1], 1.0f - ei);
    atomicAdd(&acc[2], cls);
    // tobj is binary (gr == 0): flag this cell
    atomicOr(&mask[(unsigned)(lin >> 5)], 1u << ((unsigned)lin & 31u));
}

// Bandwidth-bound pass: sum bce(p[...,4], tobj) over all cells.
// bce(l,0)=softplus(l); bce(l,1)=softplus(l)-l  -> subtract l where flagged.
// Cross-lane reduction via V_WMMA_F32_16X16X4_F32 with B = ones.
__global__ void k_obj(const float* __restrict__ p,
                      const unsigned* __restrict__ mask,
                      float* acc) {
    int tid    = blockIdx.x * blockDim.x + threadIdx.x;
    int stride = gridDim.x * blockDim.x;
    float s = 0.0f;
    for (int c = tid; c < NCELLS; c += stride) {
        // prefetch next iteration's cacheline (global_prefetch_b8)
        __builtin_prefetch(&p[(long)(c + stride) * FF + 4], 0, 0);
        float l = p[(long)c * FF + 4];
        float v = softplusf(l);
        if (mask[c >> 5] & (1u << (c & 31))) v -= l;
        s += v;
    }
    // A (16x4 f32): lane L<16 -> A[L,0..1]=(s,0); L>=16 -> A[L-16,2..3]=(s,0)
    // B (4x16) = ones  ->  D[m,n] = s_m + s_{m+16} for all n.
    v2f av; av.x = s;    av.y = 0.0f;
    v2f bv; bv.x = 1.0f; bv.y = 1.0f;
    v8f cm = {};
    cm = __builtin_amdgcn_wmma_f32_16x16x4_f32(
        /*neg_a=*/false, av, /*neg_b=*/false, bv,
        /*c_mod=*/(short)0, cm, /*reuse_a=*/false, /*reuse_b=*/false);
    // lane 0 holds M=0..7 (v=0..7), lane 16 holds M=8..15 -> together Sum over all 32 lanes
    float tsum = cm[0] + cm[1] + cm[2] + cm[3] + cm[4] + cm[5] + cm[6] + cm[7];
    unsigned lane = threadIdx.x & 31u;
    if (lane == 0u || lane == 16u) atomicAdd(&acc[3], tsum);
}

__global__ void k_final(const float* __restrict__ acc, float* __restrict__ out) {
    if (blockIdx.x == 0 && threadIdx.x == 0) {
        float nv = fmaxf(acc[0], 1.0f);
        float lb = acc[1] / nv;                                  // * 1.0
        float lc = acc[2] / (nv * (float)CC) * 10.0f;
        float lo = acc[3] / (float)NCELLS * 20.0f;
        out[0] = (lb + lo + lc) * (float)BS;
        out[1] = lo;
        out[2] = lc;
    }
}

extern "C" void kernel_launch(void* const* d_in, const int* in_sizes, int n_in,
                              void* d_out, int out_size, void* d_ws, size_t ws_size,
                              hipStream_t stream) {
    const float* p       = (const float*)d_in[0];   // (4,3,64,64,64,18) f32
    const float* targets = (const float*)d_in[1];   // (4,256,18) f32
    const float* anchor  = (const float*)d_in[2];   // (1,3,1) f32
    float*    acc  = (float*)d_ws;
    unsigned* mask = (unsigned*)((char*)d_ws + 64);
    float*    out  = (float*)d_out;

    k_zero<<<(NMASKW + 255) / 256, 256, 0, stream>>>(acc, mask);
    k_candidates<<<(NCAND + 255) / 256, 256, 0, stream>>>(p, targets, anchor, acc, mask);
    k_obj<<<3072, 256, 0, stream>>>(p, mask, acc);
    k_final<<<1, 1, 0, stream>>>(acc, out);
}